// skew_IoULoss_67637144978189
// MI455X (gfx1250) — compile-verified
//
#include <hip/hip_runtime.h>
#include <math.h>

// ---------------------------------------------------------------------------
// skew-IoU loss for rotated rectangles (Sutherland-Hodgman clip + shoelace),
// reduced to a single f32 scalar:  sum_i -log(max(iou_i, 1e-6)).
//
// Stage 1: grid-stride per-element loss, deterministic in-block tree reduce,
//          one partial per block into d_ws.
// Stage 2: single wave32 block; strided per-lane accumulation (fixed order),
//          then a V_WMMA_F32_16X16X4_F32 against an all-ones B matrix folds
//          the 32 lane accumulators; one shfl_xor finishes the scalar.
// ---------------------------------------------------------------------------

#define PI_F   3.14159265358979323846f
#define EPS_F  1e-6f
#define MAXV   8
#define NBLOCKS  512
#define NTHREADS 256

typedef __attribute__((ext_vector_type(2))) float v2f;
typedef __attribute__((ext_vector_type(8))) float v8f;

__device__ __forceinline__ void dist2rect(const float* d, float x, float y,
                                          float* vx, float* vy) {
    float t = d[0], r = d[1], b = d[2], l = d[3], th = d[4];
    th = fmodf(1.25f * PI_F + th * (PI_F * 0.5f), PI_F * 0.5f);
    if (th < PI_F * 0.25f) th += PI_F * 0.5f;
    float ct = cosf(th);
    float st = sinf(th);
    vx[0] = x - b * ct - l * st;  vy[0] = y - b * st + l * ct;
    vx[1] = x + r * st - b * ct;  vy[1] = y - r * ct - b * st;
    vx[2] = x + t * ct + r * st;  vy[2] = y + t * st - r * ct;
    vx[3] = x - l * st + t * ct;  vy[3] = y + l * ct + t * st;
}

__device__ float iou_one(const float* pd, const float* gt, float px_, float py_) {
    float area1 = (pd[0] + pd[2]) * (pd[3] + pd[1]);
    float area2 = (gt[0] + gt[2]) * (gt[3] + gt[1]);

    float r1x[4], r1y[4], r2x[4], r2y[4];
    dist2rect(pd, px_, py_, r1x, r1y);
    dist2rect(gt, px_, py_, r2x, r2y);

    float polx[MAXV], poly_[MAXV];
    int cnt = 4;
    #pragma unroll
    for (int j = 0; j < 4; ++j) { polx[j] = r1x[j]; poly_[j] = r1y[j]; }

    for (int k = 0; k < 4; ++k) {
        if (cnt <= 2) break;                       // reference breaks once <=2 verts
        int kn = (k + 1) & 3;
        // clip line a*x + b*y + c <= 0 through rect2[k] -> rect2[kn]
        float a = r2y[kn] - r2y[k];
        float b = r2x[k]  - r2x[kn];
        float c = r2x[kn] * r2y[k] - r2y[kn] * r2x[k];

        float nx[MAXV], ny[MAXV];
        int m = 0;
        for (int j = 0; j < cnt; ++j) {
            int jn = (j + 1 == cnt) ? 0 : (j + 1);
            float sx = polx[j],  sy = poly_[j];
            float tx = polx[jn], ty = poly_[jn];
            float sv = a * sx + b * sy + c;
            float tv = a * tx + b * ty + c;
            if (sv <= 0.0f) {
                if (m < MAXV) { nx[m] = sx; ny[m] = sy; }
                ++m;
            }
            if (sv * tv < 0.0f) {                  // edge crosses the clip line
                float a2 = ty - sy;
                float b2 = sx - tx;
                float c2 = tx * sy - ty * sx;
                float w  = a * b2 - b * a2;
                if (w == 0.0f) w = 1.0f;           // safe denom (masked case)
                float ix = (b * c2 - c * b2) / w;
                float iy = (c * a2 - a * c2) / w;
                if (m < MAXV) { nx[m] = ix; ny[m] = iy; }
                ++m;
            }
        }
        if (m > MAXV) m = MAXV;
        cnt = m;
        for (int j = 0; j < cnt; ++j) { polx[j] = nx[j]; poly_[j] = ny[j]; }
    }

    if (cnt <= 2) return 0.0f;

    float inter = 0.0f;
    for (int j = 0; j < cnt; ++j) {
        int jn = (j + 1 == cnt) ? 0 : (j + 1);
        inter += polx[j] * poly_[jn] - poly_[j] * polx[jn];
    }
    inter *= 0.5f;
    return inter / (area1 + area2 - inter);
}

__global__ void iou_loss_partial_kernel(const float* __restrict__ pred,
                                        const float* __restrict__ target,
                                        const float* __restrict__ pos,
                                        float* __restrict__ partial,
                                        int n) {
    __shared__ float sdata[NTHREADS];
    const int tid    = threadIdx.x;
    const int stride = gridDim.x * blockDim.x;

    float acc = 0.0f;
    for (int i = blockIdx.x * blockDim.x + tid; i < n; i += stride) {
        float pd[5], gt[5];
        #pragma unroll
        for (int j = 0; j < 5; ++j) {
            pd[j] = pred[i * 5 + j];
            gt[j] = target[i * 5 + j];
        }
        float x = pos[i * 2 + 0];
        float y = pos[i * 2 + 1];
        float iou = iou_one(pd, gt, x, y);
        float v = (iou > EPS_F) ? iou : EPS_F;     // clip below at EPS
        acc -= logf(v);
    }

    sdata[tid] = acc;
    __syncthreads();
    #pragma unroll
    for (int s = NTHREADS / 2; s > 0; s >>= 1) {   // fixed tree order: deterministic
        if (tid < s) sdata[tid] += sdata[tid + s];
        __syncthreads();
    }
    if (tid == 0) partial[blockIdx.x] = sdata[0];
}

// Single-wave final reduction. 32 lane accumulators are folded with
// V_WMMA_F32_16X16X4_F32 (B = all-ones): A[m][0] = acc_m, A[m][2] = acc_{m+16}
// => D[m][n] = acc_m + acc_{m+16}.  Lane L holds D[0..7][L] (L<16) or
// D[8..15][L-16] (L>=16); lane-local 8-sum + shfl_xor(16) yields the total.
__global__ void final_reduce_wmma_kernel(const float* __restrict__ partial,
                                         float* __restrict__ out,
                                         int npart) {
    const int lane = threadIdx.x;                  // blockDim.x == 32, one wave
    float acc = 0.0f;
    for (int j = lane; j < npart; j += 32)         // fixed order per lane
        acc += partial[j];

    v2f a; a.x = acc;  a.y = 0.0f;                 // A: 16x4, acc in K=0 / K=2 slots
    v2f b; b.x = 1.0f; b.y = 1.0f;                 // B: 4x16 all-ones
    v8f c = {};
    // 8 args: (neg_a, A, neg_b, B, c_mod, C, reuse_a, reuse_b)
    v8f d = __builtin_amdgcn_wmma_f32_16x16x4_f32(
        false, a, false, b, (short)0, c, false, false);

    float s = d[0] + d[1] + d[2] + d[3] + d[4] + d[5] + d[6] + d[7];
    s += __shfl_xor(s, 16, 32);                    // pair lanes L and L^16
    if (lane == 0) out[0] = s;                     // overwrite: no init needed
}

extern "C" void kernel_launch(void* const* d_in, const int* in_sizes, int n_in,
                              void* d_out, int out_size, void* d_ws, size_t ws_size,
                              hipStream_t stream) {
    const float* pred   = (const float*)d_in[0];   // [N,5]
    const float* target = (const float*)d_in[1];   // [N,5]
    const float* pos    = (const float*)d_in[2];   // [N,2]
    const int n = in_sizes[0] / 5;

    float* partial = (float*)d_ws;                 // NBLOCKS floats (2 KB)

    iou_loss_partial_kernel<<<NBLOCKS, NTHREADS, 0, stream>>>(
        pred, target, pos, partial, n);
    final_reduce_wmma_kernel<<<1, 32, 0, stream>>>(
        partial, (float*)d_out, NBLOCKS);
}